// HardNegativePointLoss_1752346657499
// MI455X (gfx1250) — compile-verified
//
#include <hip/hip_runtime.h>

typedef __attribute__((ext_vector_type(2))) float v2f;
typedef __attribute__((ext_vector_type(8))) float v8f;
typedef unsigned int u32;
typedef u32 u32x4 __attribute__((ext_vector_type(4)));
typedef int  i32x4 __attribute__((ext_vector_type(4)));
typedef int  i32x8 __attribute__((ext_vector_type(8)));

#define TINV      (1.0f / 0.07f)
#define KTOP      4096
#define NBINS     2048
#define BIN_LO    (-8.0f)
#define BIN_SCALE (128.0f)
#define EPS       (1e-7f)
#define LDS_STR   132          // padded row stride (DWORDs): 132 % 64 = 4 -> conflict-free
#define TILES_PER_BLOCK 4      // column tiles (128 cols each) per block

#if defined(__has_builtin)
#if __has_builtin(__builtin_amdgcn_tensor_load_to_lds) && __has_builtin(__builtin_amdgcn_s_wait_tensorcnt)
#define HAVE_TDM 1
#endif
#endif

#if HAVE_TDM
#warning "CDNA5 diag: TDM path ACTIVE (__builtin_amdgcn_tensor_load_to_lds found)"
#if defined(__clang_major__) && __clang_major__ >= 23
#warning "CDNA5 diag: TDM 6-arg arity (clang >= 23)"
#else
#warning "CDNA5 diag: TDM 5-arg arity (clang < 23)"
#endif
#else
#warning "CDNA5 diag: TDM builtin NOT found - cooperative float4 staging fallback"
#endif

// ---------------------------------------------------------------------------
// K0: L2-normalize rows of points -> normP  (B blocks x D threads, D=128)
// ---------------------------------------------------------------------------
__global__ void k_normalize(const float* __restrict__ pts,
                            float* __restrict__ normP, int D) {
  __shared__ float red[128];
  const int row = blockIdx.x;
  const float v = pts[(size_t)row * D + threadIdx.x];
  red[threadIdx.x] = v * v;
  __syncthreads();
  for (int s = 64; s > 0; s >>= 1) {
    if ((int)threadIdx.x < s) red[threadIdx.x] += red[threadIdx.x + s];
    __syncthreads();
  }
  normP[(size_t)row * D + threadIdx.x] = v / sqrtf(red[0]);
}

#if HAVE_TDM
// Issue one 2D TDM load: 128 f32 x `rows` rows of `bank` tile -> LDS, with
// hardware row padding to a 132-DWORD stride (pad_interval=128 DW, amount=4 DW).
__device__ __forceinline__ void tdm_issue_tile(const float* gptr, u32 ldsAddr, u32 rowsRem) {
  const unsigned long long ga = (unsigned long long)(const void*)gptr;
  u32x4 g0;
  g0[0] = 1u;                                        // count=1, user descriptor
  g0[1] = ldsAddr;                                   // lds_addr (bytes)
  g0[2] = (u32)ga;                                   // global_addr[31:0]
  g0[3] = (u32)((ga >> 32) & 0x01FFFFFFull) | (2u << 30);  // addr[56:32] | type=2

  i32x8 g1;
  g1[0] = (int)((2u << 16)                           // data_size = 4B
              | (1u << 20)                           // pad_enable
              | (6u << 22)                           // pad_interval: 128 DWORDs
              | (3u << 25));                         // pad_amount:   4 DWORDs
  g1[1] = (int)(128u << 16);                         // tensor_dim0 = 128 elems
  g1[2] = (int)((rowsRem & 0xFFFFu) << 16);          // tensor_dim1[15:0]
  g1[3] = (int)((rowsRem >> 16) | (128u << 16));     // tensor_dim1[31:16] | tile_dim0=128
  g1[4] = 128;                                       // tile_dim1 = 128 rows
  g1[5] = 128;                                       // tensor_dim0_stride = 128 elems
  g1[6] = 0;
  g1[7] = 0;

  const i32x4 z4 = {0, 0, 0, 0};
#if defined(__clang_major__) && __clang_major__ >= 23
  const i32x8 z8 = {0, 0, 0, 0, 0, 0, 0, 0};
  __builtin_amdgcn_tensor_load_to_lds(g0, g1, z4, z4, z8, 0);
#else
  __builtin_amdgcn_tensor_load_to_lds(g0, g1, z4, z4, 0);
#endif
}
#endif

// ---------------------------------------------------------------------------
// K1: similarities = normP @ bank^T via V_WMMA_F32_16X16X4_F32.
// Block = 256 threads (8 waves): 16 batch rows x (4 tiles x 128 bank cols).
// B tiles double-buffered in LDS via the Tensor Data Mover (issue t+1, then
// s_wait_tensorcnt(1) guarantees tile t landed). A fragments held in VGPRs.
// Each wave: per tile, 16x ds_load_2addr_b64 + 32 chained f32 WMMAs (two
// independent accumulator chains).
// ---------------------------------------------------------------------------
__global__ void __launch_bounds__(256) k_gemm(const float* __restrict__ normP,
                                              const float* __restrict__ bank,
                                              float* __restrict__ simOut, int N) {
  __shared__ __align__(16) float sB[2 * 128 * LDS_STR];   // double buffer, 132KB

  const int m0 = blockIdx.y * 16;
  const int n0 = blockIdx.x * (128 * TILES_PER_BLOCK);
  const int Tcnt = min(TILES_PER_BLOCK, (N - n0 + 127) >> 7);

  const int lane = threadIdx.x & 31;
  const int wave = threadIdx.x >> 5;
  const int half = lane >> 4;                 // 0: lanes 0-15, 1: lanes 16-31
  const int mrow = lane & 15;

  // A fragments for this wave's 16x16 tile row, held in registers (32 x v2f).
  const float* __restrict__ aG = normP + (size_t)(m0 + mrow) * 128 + 2 * half;
  v2f areg[32];
#pragma unroll
  for (int k4 = 0; k4 < 32; ++k4) areg[k4] = *(const v2f*)(aG + k4 * 4);

#if HAVE_TDM
  if (threadIdx.x < 32)                        // wave 0 issues tile 0
    tdm_issue_tile(bank + (size_t)n0 * 128,
                   (u32)(unsigned long long)(void*)sB, (u32)(N - n0));
#endif

  for (int t = 0; t < Tcnt; ++t) {
    const int nt0 = n0 + t * 128;
    float* __restrict__ sBt = sB + (t & 1) * (128 * LDS_STR);

    __syncthreads();   // waves done reading buffer (t+1)&1 (used by tile t-1)

#if HAVE_TDM
    if (t + 1 < Tcnt) {
      if (threadIdx.x < 32)
        tdm_issue_tile(bank + (size_t)(nt0 + 128) * 128,
                       (u32)(unsigned long long)(void*)(sB + ((t + 1) & 1) * (128 * LDS_STR)),
                       (u32)(N - (nt0 + 128)));
      __builtin_amdgcn_s_wait_tensorcnt(1);    // tile t complete (in-order TDM)
    } else {
      __builtin_amdgcn_s_wait_tensorcnt(0);
    }
#else
    for (int id = threadIdx.x; id < 128 * 32; id += 256) {
      const int r = id >> 5, ck = id & 31;
      int nr = nt0 + r;
      nr = nr < N ? nr : N - 1;
      *(float4*)(sBt + r * LDS_STR + ck * 4) =
          *(const float4*)(bank + (size_t)nr * 128 + ck * 4);
    }
#endif
    __syncthreads();   // publish tile t's LDS data

    const float* __restrict__ bptr = sBt + (wave * 16 + mrow) * LDS_STR + 2 * half;

    v8f c0 = {}, c1 = {};
#pragma unroll
    for (int kk = 0; kk < 128; kk += 8) {
      const v2f a0 = areg[kk >> 2];
      const v2f a1 = areg[(kk >> 2) + 1];
      const v2f b0 = *(const v2f*)(bptr + kk);
      const v2f b1 = *(const v2f*)(bptr + kk + 4);
      c0 = __builtin_amdgcn_wmma_f32_16x16x4_f32(false, a0, false, b0,
                                                 (short)0, c0, false, false);
      c1 = __builtin_amdgcn_wmma_f32_16x16x4_f32(false, a1, false, b1,
                                                 (short)0, c1, false, false);
    }

    const int n = nt0 + wave * 16 + mrow;      // bank column this lane owns
    if (n < N) {
#pragma unroll
      for (int j = 0; j < 8; ++j) {
        const int m = m0 + j + 8 * half;
        simOut[(size_t)m * N + n] = c0[j] + c1[j];
      }
    }
  }
}

// ---------------------------------------------------------------------------
// K2: per-row histogram of similarities (one block per row), float4 stream
// ---------------------------------------------------------------------------
__global__ void k_hist(const float* __restrict__ sim,
                       int* __restrict__ hist, int N) {
  __shared__ int h[NBINS];
  const int row = blockIdx.x;
  for (int i = threadIdx.x; i < NBINS; i += blockDim.x) h[i] = 0;
  __syncthreads();
  const float4* __restrict__ s4 = (const float4*)(sim + (size_t)row * N);
  const int n4 = N >> 2;
  for (int i = threadIdx.x; i < n4; i += blockDim.x) {
    const float4 v = s4[i];
    const float vv[4] = {v.x, v.y, v.z, v.w};
#pragma unroll
    for (int j = 0; j < 4; ++j) {
      int b = (int)((vv[j] - BIN_LO) * BIN_SCALE);
      b = b < 0 ? 0 : (b > NBINS - 1 ? NBINS - 1 : b);
      atomicAdd(&h[b], 1);
    }
  }
  for (int i = (n4 << 2) + threadIdx.x; i < N; i += blockDim.x) {
    int b = (int)((sim[(size_t)row * N + i] - BIN_LO) * BIN_SCALE);
    b = b < 0 ? 0 : (b > NBINS - 1 ? NBINS - 1 : b);
    atomicAdd(&h[b], 1);
  }
  __syncthreads();
  for (int i = threadIdx.x; i < NBINS; i += blockDim.x)
    hist[(size_t)row * NBINS + i] = h[i];
}

// ---------------------------------------------------------------------------
// K3: per-row threshold bin (K-th largest in thrBin; thrAbove strictly above)
// ---------------------------------------------------------------------------
__global__ void k_thresh(const int* __restrict__ hist,
                         int* __restrict__ thrBin, int* __restrict__ thrAbove) {
  const int row = threadIdx.x;
  const int* __restrict__ h = hist + (size_t)row * NBINS;
  int cnt = 0, b = NBINS - 1;
  for (; b >= 0; --b) {
    const int c = h[b];
    if (cnt + c >= KTOP) break;
    cnt += c;
  }
  if (b < 0) b = 0;
  thrBin[row] = b;
  thrAbove[row] = cnt;
}

// ---------------------------------------------------------------------------
// K4: per-row denominator (sum of exp over top-K) + per-row loss term
// ---------------------------------------------------------------------------
__global__ void k_denom(const float* __restrict__ sim,
                        const int* __restrict__ thrBin,
                        const int* __restrict__ thrAbove,
                        const int* __restrict__ idx,
                        float* __restrict__ rowloss, int N) {
  __shared__ float rA[1024];
  __shared__ float rB[1024];
  __shared__ int   rC[1024];
  const int row = blockIdx.x;
  const int tid = threadIdx.x;
  const float* __restrict__ s = sim + (size_t)row * N;
  const int bth = thrBin[row];

  float acc = 0.f, accB = 0.f;
  int cB = 0;
  const float4* __restrict__ s4 = (const float4*)s;
  const int n4 = N >> 2;
  for (int i = tid; i < n4; i += blockDim.x) {
    const float4 v4 = s4[i];
    const float vv[4] = {v4.x, v4.y, v4.z, v4.w};
#pragma unroll
    for (int j = 0; j < 4; ++j) {
      const float v = vv[j];
      int b = (int)((v - BIN_LO) * BIN_SCALE);
      b = b < 0 ? 0 : (b > NBINS - 1 ? NBINS - 1 : b);
      if (b >= bth) {
        const float e = __expf(v * TINV);
        if (b > bth) acc += e;
        else { accB += e; ++cB; }
      }
    }
  }
  for (int i = (n4 << 2) + tid; i < N; i += blockDim.x) {
    const float v = s[i];
    int b = (int)((v - BIN_LO) * BIN_SCALE);
    b = b < 0 ? 0 : (b > NBINS - 1 ? NBINS - 1 : b);
    if (b >= bth) {
      const float e = __expf(v * TINV);
      if (b > bth) acc += e;
      else { accB += e; ++cB; }
    }
  }
  rA[tid] = acc; rB[tid] = accB; rC[tid] = cB;
  __syncthreads();
  for (int st = (int)blockDim.x >> 1; st > 0; st >>= 1) {
    if (tid < st) { rA[tid] += rA[tid + st]; rB[tid] += rB[tid + st]; rC[tid] += rC[tid + st]; }
    __syncthreads();
  }
  if (tid == 0) {
    const float need  = (float)(KTOP - thrAbove[row]);   // taken from boundary bin
    const float meanB = rB[0] / fmaxf((float)rC[0], 1.0f);
    const float denom = rA[0] + need * meanB;
    const float spos  = s[idx[row]];
    const float p     = __expf(spos * TINV);
    rowloss[row] = -__logf(p / denom + EPS);
  }
}

// ---------------------------------------------------------------------------
// K5: loss = mean(rowloss) -> d_out[0]
// ---------------------------------------------------------------------------
__global__ void k_loss(const float* __restrict__ rowloss, float* __restrict__ out, int B) {
  __shared__ float r[256];
  r[threadIdx.x] = rowloss[threadIdx.x];
  __syncthreads();
  for (int st = B >> 1; st > 0; st >>= 1) {
    if ((int)threadIdx.x < st) r[threadIdx.x] += r[threadIdx.x + st];
    __syncthreads();
  }
  if (threadIdx.x == 0) out[0] = r[0] / (float)B;
}

// ---------------------------------------------------------------------------
extern "C" void kernel_launch(void* const* d_in, const int* in_sizes, int n_in,
                              void* d_out, int out_size, void* d_ws, size_t ws_size,
                              hipStream_t stream) {
  const float* points = (const float*)d_in[0];
  const int*   pidx   = (const int*)d_in[1];
  const float* bank   = (const float*)d_in[2];

  const int B = in_sizes[1];            // 256
  const int D = in_sizes[0] / B;        // 128
  const int N = in_sizes[2] / D;        // 500000

  float* out    = (float*)d_out;
  float* simOut = out + 1;              // output order: (loss, similarities)

  char*  ws       = (char*)d_ws;
  float* normP    = (float*)ws;                                   // B*D f32
  int*   hist     = (int*)(ws + sizeof(float) * (size_t)B * D);   // B*NBINS i32
  int*   thrBin   = (int*)((char*)hist + sizeof(int) * (size_t)B * NBINS);
  int*   thrAbove = thrBin + B;
  float* rowloss  = (float*)(thrAbove + B);

  k_normalize<<<B, 128, 0, stream>>>(points, normP, D);

  const int colsPerBlock = 128 * TILES_PER_BLOCK;
  dim3 g((N + colsPerBlock - 1) / colsPerBlock, B / 16);
  k_gemm<<<g, 256, 0, stream>>>(normP, bank, simOut, N);

  k_hist<<<B, 1024, 0, stream>>>(simOut, hist, N);
  k_thresh<<<1, B, 0, stream>>>(hist, thrBin, thrAbove);
  k_denom<<<B, 1024, 0, stream>>>(simOut, thrBin, thrAbove, pidx, rowloss, N);
  k_loss<<<1, B, 0, stream>>>(rowloss, out, B);
}